// Sampler_21577915695192
// MI455X (gfx1250) — compile-verified
//
#include <hip/hip_runtime.h>
#include <stdint.h>

// ---------------------------------------------------------------------------
// Sampler: tokens[b] = argmax_v( softmax(logits[b]/T[b])[v] / exp_noise[b,v] )
//                    = argmax_v( logits[b,v]/T[b] - log(max(noise,1e-10)) )
//   (softmax is a strictly monotone per-row map; log is monotone -> argmax
//    is preserved, collapsing 3 passes into one streaming pass.)
// Memory-bound: ~131 MB -> ~5.6 us floor @ 23.3 TB/s.
// Kernel 1: grid (NSPLIT x B) = 5 x 128 blocks, each an 8-deep
//           ASYNCcnt-pipelined streaming argmax over 25600 elements
//           (global_load_async_to_lds_b128 staging, wave32 shfl reduce).
// Kernel 2: fold the 5 partials per row (ascending -> first-max tie rule).
// ---------------------------------------------------------------------------

#define TPB    256               // 8 wave32 waves per block
#define PD     8                 // async pipeline depth (buffers)
#define VEC    4                 // floats per lane per stage (b128)
#define CHUNK  (TPB * VEC)       // 1024 floats per block-iteration per array
#define NSPLIT 5                 // segments per row: 128000 = 5 * 25 * 1024

typedef float  float4v __attribute__((ext_vector_type(4)));
typedef const void __attribute__((address_space(3))) lds_cvoid;

// LDS byte offset of a __shared__ object (generic -> AS3 -> integer).
__device__ __forceinline__ uint32_t lds_off(const void* p) {
    return (uint32_t)(uintptr_t)(lds_cvoid*)p;
}

// One 16B-per-lane async copy global -> LDS (GV mode, 64-bit vaddr).
// Tracked by ASYNCcnt; no VGPR destination, no LOADcnt pressure.
__device__ __forceinline__ void async_ld16(uint32_t lds_byte, const void* gaddr) {
    asm volatile("global_load_async_to_lds_b128 %0, %1, off"
                 :
                 : "v"(lds_byte), "v"((uint64_t)(uintptr_t)gaddr)
                 : "memory");
}

// Steady-state: 2 async ops per iteration, PD buffers -> allow 2*(PD-1)=14.
__device__ __forceinline__ void wait_async_le14() {
    asm volatile("s_wait_asynccnt 14" ::: "memory");
}
__device__ __forceinline__ void wait_async_0() {
    asm volatile("s_wait_asynccnt 0" ::: "memory");
}

__global__ __launch_bounds__(TPB) void sampler_partial_kernel(
    const float* __restrict__ logits,        // [B, V]
    const float* __restrict__ temperatures,  // [B]
    const float* __restrict__ exp_noise,     // [B, V]
    float* __restrict__ part_v,              // [B * NSPLIT]
    int*   __restrict__ part_i,              // [B * NSPLIT]
    int V, int iters)                        // iters = V/(NSPLIT*CHUNK) >= PD
{
    __shared__ float4v stageL[PD][TPB];      // 32 KB
    __shared__ float4v stageN[PD][TPB];      // 32 KB
    __shared__ float   red_v[TPB / 32];
    __shared__ int     red_i[TPB / 32];

    const int seg = blockIdx.x;              // 0..NSPLIT-1
    const int row = blockIdx.y;              // 0..B-1
    const int tid = threadIdx.x;

    const int segBase = seg * iters * CHUNK; // row-local element offset

    const float invT = 1.0f / temperatures[row];
    const float* __restrict__ rowL = logits    + (size_t)row * (size_t)V + segBase;
    const float* __restrict__ rowN = exp_noise + (size_t)row * (size_t)V + segBase;

    // ---- prologue: fill the pipeline (iters >= PD guaranteed by host) ----
#pragma unroll
    for (int p = 0; p < PD; ++p) {
        const int base = p * CHUNK + tid * VEC;
        async_ld16(lds_off(&stageL[p][tid]), rowL + base);
        async_ld16(lds_off(&stageN[p][tid]), rowN + base);
    }

    float bv = -__builtin_inff();
    int   bi = 0;

    // ---- main software-pipelined loop -----------------------------------
    const int mainIters = iters - PD;
    for (int it = 0; it < mainIters; ++it) {
        wait_async_le14();                   // iteration `it` data now in LDS
        const int b = it & (PD - 1);
        const float4v lg = stageL[b][tid];   // ds_load_b128
        const float4v nz = stageN[b][tid];   // ds_load_b128
        const int idx0 = segBase + it * CHUNK + tid * VEC;
#pragma unroll
        for (int j = 0; j < VEC; ++j) {
            const float key = lg[j] * invT - __logf(fmaxf(nz[j], 1e-10f));
            if (key > bv) { bv = key; bi = idx0 + j; }   // ascending -> first-max
        }
        const int nxt  = it + PD;
        const int nb   = nxt & (PD - 1);
        const int base = nxt * CHUNK + tid * VEC;
        async_ld16(lds_off(&stageL[nb][tid]), rowL + base);
        async_ld16(lds_off(&stageN[nb][tid]), rowN + base);
    }

    // ---- drain: everything issued, wait once, consume the rest ----------
    wait_async_0();
    for (int it = mainIters; it < iters; ++it) {
        const int b = it & (PD - 1);
        const float4v lg = stageL[b][tid];
        const float4v nz = stageN[b][tid];
        const int idx0 = segBase + it * CHUNK + tid * VEC;
#pragma unroll
        for (int j = 0; j < VEC; ++j) {
            const float key = lg[j] * invT - __logf(fmaxf(nz[j], 1e-10f));
            if (key > bv) { bv = key; bi = idx0 + j; }
        }
    }

    // ---- wave32 butterfly argmax (tie -> lowest index) ------------------
#pragma unroll
    for (int m = 16; m >= 1; m >>= 1) {
        const float ov = __shfl_xor(bv, m, 32);
        const int   oi = __shfl_xor(bi, m, 32);
        if (ov > bv || (ov == bv && oi < bi)) { bv = ov; bi = oi; }
    }
    const int wid = tid >> 5;
    if ((tid & 31) == 0) { red_v[wid] = bv; red_i[wid] = bi; }
    __syncthreads();

    if (tid == 0) {
        float fv = red_v[0];
        int   fi = red_i[0];
#pragma unroll
        for (int w = 1; w < TPB / 32; ++w) {
            const float wv = red_v[w];
            const int   wi = red_i[w];
            if (wv > fv || (wv == fv && wi < fi)) { fv = wv; fi = wi; }
        }
        part_v[row * NSPLIT + seg] = fv;
        part_i[row * NSPLIT + seg] = fi;
    }
}

// Fold NSPLIT partials per row. Segments ascend in index, so scanning
// ascending with strict '>' reproduces jnp.argmax's first-index tie rule.
__global__ void sampler_finalize_kernel(const float* __restrict__ part_v,
                                        const int*   __restrict__ part_i,
                                        int* __restrict__ tokens, int B)
{
    const int b = blockIdx.x * blockDim.x + threadIdx.x;
    if (b >= B) return;
    float fv = -__builtin_inff();
    int   fi = 0;
#pragma unroll
    for (int s = 0; s < NSPLIT; ++s) {
        const float v = part_v[b * NSPLIT + s];
        const int   i = part_i[b * NSPLIT + s];
        if (v > fv || (v == fv && i < fi)) { fv = v; fi = i; }
    }
    tokens[b] = fi;
}

extern "C" void kernel_launch(void* const* d_in, const int* in_sizes, int n_in,
                              void* d_out, int out_size, void* d_ws, size_t ws_size,
                              hipStream_t stream) {
    const float* logits = (const float*)d_in[0];   // [B, V] f32
    const float* temps  = (const float*)d_in[1];   // [B]    f32
    const float* noise  = (const float*)d_in[2];   // [B, V] f32

    const int B = in_sizes[1];                     // 128
    const int V = in_sizes[0] / B;                 // 128000
    const int iters = V / (NSPLIT * CHUNK);        // 25 for V=128000

    float* part_v = (float*)d_ws;                  // [B*NSPLIT]
    int*   part_i = (int*)(part_v + (size_t)B * NSPLIT);

    sampler_partial_kernel<<<dim3(NSPLIT, B), dim3(TPB), 0, stream>>>(
        logits, temps, noise, part_v, part_i, V, iters);

    sampler_finalize_kernel<<<dim3((B + 127) / 128), dim3(128), 0, stream>>>(
        part_v, part_i, (int*)d_out, B);
}